// Equivariant_257698037971
// MI455X (gfx1250) — compile-verified
//
#include <hip/hip_runtime.h>

typedef __attribute__((ext_vector_type(2))) float v2f;
typedef __attribute__((ext_vector_type(4))) float v4f;
typedef __attribute__((ext_vector_type(8))) float v8f;

#define M_DIM 512
#define F_DIM 64
#define O_DIM 128
#define XS_STRIDE 68   // rows 16B-aligned AND bank-clean (4r+c mod 64 distinct)

// Barrier that orders only LDS traffic: skips the s_wait_storecnt 0 that
// __syncthreads() inserts (our global stores are never read by anyone).
__device__ __forceinline__ void lds_barrier() {
    asm volatile("s_wait_dscnt 0x0\n\t"
                 "s_barrier_signal -1\n\t"
                 "s_barrier_wait -1" ::: "memory");
}

__global__ __launch_bounds__(256)
void equivariant_kernel(const float* __restrict__ x,
                        const float* __restrict__ lam,
                        const float* __restrict__ gam,
                        float* __restrict__ out) {
    __shared__ float redS[16 * 64];
    __shared__ float colS[64];
    __shared__ float poolS[O_DIM];
    __shared__ float xS[2][16 * XS_STRIDE];   // double-buffered x tile

    const int tid  = threadIdx.x;
    const int b    = blockIdx.x;
    const int lane = tid & 31;
    const int wave = tid >> 5;       // 0..7 -> O-tile index
    const int ln16 = lane & 15;
    const int hi   = lane >> 4;      // 0 or 1 (upper half-wave)

    const size_t xb = (size_t)b * (M_DIM * F_DIM);
    const size_t ob = (size_t)b * ((size_t)M_DIM * O_DIM);

    // ---- B (lam) fragments for this wave's fixed 16-wide O-tile; loaded once ----
    // B 4x16 layout: VGPR0 = {K=0 lanes 0-15, K=2 lanes 16-31}, VGPR1 = {K=1, K=3}
    v2f bfrag[16];
    {
        const int n  = wave * 16 + ln16;
        const int kb = hi * 2;
        #pragma unroll
        for (int kc = 0; kc < 16; ++kc) {
            const int k0 = kc * 4 + kb;
            v2f bb;
            bb.x = lam[(size_t)k0 * O_DIM + n];
            bb.y = lam[(size_t)(k0 + 1) * O_DIM + n];
            bfrag[kc] = bb;
        }
    }

    // ---- Phase 1: colsum[f] = sum_m x[b,m,f]  (vectorized b128 streaming) ----
    {
        const int f4 = (tid & 15) * 4;   // 4 consecutive features
        const int g  = tid >> 4;         // 0..15 row group
        v4f s = {};
        for (int m = g; m < M_DIM; m += 16)
            s += *(const v4f*)(x + xb + (size_t)m * F_DIM + f4);
        *(v4f*)(&redS[g * 64 + f4]) = s;
    }
    lds_barrier();
    if (tid < 64) {
        float s = 0.f;
        #pragma unroll
        for (int g = 0; g < 16; ++g)
            s += redS[g * 64 + tid];
        colS[tid] = s;
    }
    lds_barrier();
    // ---- pooled[o] = colsum . gam[:,o] ----
    if (tid < O_DIM) {
        float p = 0.f;
        #pragma unroll
        for (int f = 0; f < F_DIM; ++f)
            p += colS[f] * gam[(size_t)f * O_DIM + tid];
        poolS[tid] = p;
    }
    lds_barrier();

    // per-lane pooled value for this wave's O-tile column (N = ln16)
    const float pv = poolS[wave * 16 + ln16];

    // staging coordinates (all 256 threads cover one 16x64 tile, coalesced b128)
    const int srow = tid >> 4;
    const int scol = (tid & 15) * 4;
    const int sofs = srow * XS_STRIDE + scol;

    // A-fragment LDS base: row = ln16, K sub-offset = hi*2
    const int aBase = ln16 * XS_STRIDE + hi * 2;

    // prologue: stage tile 0 into buffer 0
    {
        v4f v = *(const v4f*)(x + xb + (size_t)srow * F_DIM + scol);
        *(v4f*)(&xS[0][sofs]) = v;
    }

    // ---- Phase 2: GEMM over 32 M-tiles, double-buffered, 1 barrier/tile ----
    for (int mt = 0; mt < 32; ++mt) {
        lds_barrier();                 // buf[mt&1] staged; prior reads of other buf done
        const int cur = mt & 1;

        // issue next tile's global load early; latency hidden by 16 WMMAs
        v4f nxt;
        const bool more = (mt + 1) < 32;
        if (more)
            nxt = *(const v4f*)(x + xb + (size_t)((mt + 1) * 16 + srow) * F_DIM + scol);

        // A 16x4 layout: VGPR0 = {K=0 | K=2}, VGPR1 = {K=1 | K=3} per half-wave
        v2f afrag[16];
        #pragma unroll
        for (int kc = 0; kc < 16; ++kc)
            afrag[kc] = *(const v2f*)(&xS[cur][aBase + kc * 4]);  // ds_load_b64

        v8f acc = {};
        #pragma unroll
        for (int kc = 0; kc < 16; ++kc)
            acc = __builtin_amdgcn_wmma_f32_16x16x4_f32(
                      /*neg_a=*/false, afrag[kc],
                      /*neg_b=*/false, bfrag[kc],
                      /*c_mod=*/(short)0, acc,
                      /*reuse_a=*/false, /*reuse_b=*/false);

        // epilogue: C/D layout lane l: N = l%16, M = vgpr + 8*(l>=16)
        {
            const int n = wave * 16 + ln16;
            #pragma unroll
            for (int i = 0; i < 8; ++i) {
                const int m = mt * 16 + i + hi * 8;
                const float r = acc[i] - pv;
                out[ob + (size_t)m * O_DIM + n] = r > 0.f ? r : 0.f;
            }
        }

        // stage next tile into the other buffer (consumed after next barrier)
        if (more)
            *(v4f*)(&xS[cur ^ 1][sofs]) = nxt;
    }
}

extern "C" void kernel_launch(void* const* d_in, const int* in_sizes, int n_in,
                              void* d_out, int out_size, void* d_ws, size_t ws_size,
                              hipStream_t stream) {
    const float* x   = (const float*)d_in[0];
    const float* lam = (const float*)d_in[1];
    const float* gam = (const float*)d_in[2];
    float* out = (float*)d_out;

    dim3 grid(8192);   // one workgroup per batch b
    dim3 block(256);   // 8 wave32
    equivariant_kernel<<<grid, block, 0, stream>>>(x, lam, gam, out);
}